// SAGE_4973572128780
// MI455X (gfx1250) — compile-verified
//
#include <hip/hip_runtime.h>
#include <hip/hip_bf16.h>

typedef __attribute__((ext_vector_type(2))) float v2f;
typedef __attribute__((ext_vector_type(8))) float v8f;

// Problem constants (fixed by the reference harness)
constexpr int IN  = 128;
constexpr int HID = 256;
constexpr int OUT = 64;
constexpr int N1  = 40000;
constexpr int N2  = 4000;

// ---------------------------------------------------------------------------
// Edge scatter: one wave32 per edge. Each lane moves F/32 floats (float4
// chunks) from source row to target accumulator row via f32 atomics (resolve
// in L2: agg buffers are far smaller than the 192MB L2). Lane 0 bumps degree.
// ---------------------------------------------------------------------------
template<int F>
__global__ void sage_scatter(const float* __restrict__ xs,
                             const int* __restrict__ src,
                             const int* __restrict__ dst,
                             float* __restrict__ agg,
                             float* __restrict__ deg,
                             int E) {
    int gid  = blockIdx.x * blockDim.x + threadIdx.x;
    int e    = gid >> 5;
    int lane = gid & 31;
    if (e >= E) return;
    int s = src[e];
    int d = dst[e];
    const float* xr = xs + (size_t)s * F;
    float*       ar = agg + (size_t)d * F;
#pragma unroll
    for (int c = 0; c < F / 128; ++c) {
        int off = c * 128 + lane * 4;
        float4 v = *(const float4*)(xr + off);
        atomicAdd(ar + off + 0, v.x);
        atomicAdd(ar + off + 1, v.y);
        atomicAdd(ar + off + 2, v.z);
        atomicAdd(ar + off + 3, v.w);
    }
    if (lane == 0) atomicAdd(deg + d, 1.0f);
}

// ---------------------------------------------------------------------------
// Row-wise mean normalization: agg[i] /= max(deg[row],1). cols = 1<<shift.
// ---------------------------------------------------------------------------
__global__ void sage_norm(float* __restrict__ agg,
                          const float* __restrict__ deg,
                          int total, int shift) {
    int i = blockIdx.x * blockDim.x + threadIdx.x;
    if (i >= total) return;
    float dv = deg[i >> shift];
    agg[i] *= 1.0f / fmaxf(dv, 1.0f);
}

// ---------------------------------------------------------------------------
// Fused dual GEMM via V_WMMA_F32_16X16X4_F32 (wave32 matrix core):
//   out[16 x 16*NT] = A1_tile @ W1 + A2_tile @ W2 + bias  (optional ReLU)
// One wave owns a 16x(16*NT) output tile; the A fragment (2 f32/lane) is
// reused across the NT accumulators per K-step.
// A-frag layout (ISA 7.12.2): lane m = lane&15, K-base = (lane>>4)*2, 2 K vals.
// B-frag symmetric with N on lanes; C/D: 8 VGPRs, M = v + (lane>>4)*8.
// ---------------------------------------------------------------------------
template<int K, int N, int NT, bool RELU>
__global__ void sage_gemm(const float* __restrict__ A1,
                          const float* __restrict__ A2,
                          const float* __restrict__ W1,
                          const float* __restrict__ W2,
                          const float* __restrict__ bias,
                          float* __restrict__ out,
                          int Mtiles) {
    constexpr int NTILES = N / (16 * NT);
    int wid  = (blockIdx.x * blockDim.x + threadIdx.x) >> 5;
    int lane = threadIdx.x & 31;
    int tm = wid / NTILES;
    int tn = wid % NTILES;
    if (tm >= Mtiles) return;

    int row0 = tm * 16;
    int col0 = tn * 16 * NT;
    int m    = lane & 15;
    int kb   = (lane >> 4) * 2;   // K sub-offset for A/B fragments
    int n    = lane & 15;

    v8f acc[NT] = {};

    const float* a1 = A1 + (size_t)(row0 + m) * K + kb;
    const float* a2 = A2 + (size_t)(row0 + m) * K + kb;

    for (int k = 0; k < K; k += 4) {
        v2f a = *(const v2f*)(a1 + k);
#pragma unroll
        for (int t = 0; t < NT; ++t) {
            v2f b;
            b.x = W1[(size_t)(k + kb)     * N + col0 + t * 16 + n];
            b.y = W1[(size_t)(k + kb + 1) * N + col0 + t * 16 + n];
            acc[t] = __builtin_amdgcn_wmma_f32_16x16x4_f32(
                false, a, false, b, (short)0, acc[t], false, false);
        }
    }
    for (int k = 0; k < K; k += 4) {
        v2f a = *(const v2f*)(a2 + k);
#pragma unroll
        for (int t = 0; t < NT; ++t) {
            v2f b;
            b.x = W2[(size_t)(k + kb)     * N + col0 + t * 16 + n];
            b.y = W2[(size_t)(k + kb + 1) * N + col0 + t * 16 + n];
            acc[t] = __builtin_amdgcn_wmma_f32_16x16x4_f32(
                false, a, false, b, (short)0, acc[t], false, false);
        }
    }

    int mb = (lane >> 4) * 8;
#pragma unroll
    for (int t = 0; t < NT; ++t) {
        float bv = bias[col0 + t * 16 + n];
#pragma unroll
        for (int v = 0; v < 8; ++v) {
            float val = acc[t][v] + bv;
            if (RELU) val = fmaxf(val, 0.0f);
            out[(size_t)(row0 + mb + v) * N + col0 + t * 16 + n] = val;
        }
    }
}

// ---------------------------------------------------------------------------
// log_softmax over 64 classes: one wave32 per row, 2 elems/lane, shfl-xor
// reductions for max and sum(exp).
// ---------------------------------------------------------------------------
__global__ void sage_logsoftmax(float* __restrict__ out, int rows) {
    int gid  = blockIdx.x * blockDim.x + threadIdx.x;
    int row  = gid >> 5;
    int lane = gid & 31;
    if (row >= rows) return;
    float* p = out + (size_t)row * OUT;
    float a = p[lane];
    float b = p[lane + 32];
    float mx = fmaxf(a, b);
#pragma unroll
    for (int off = 16; off > 0; off >>= 1)
        mx = fmaxf(mx, __shfl_xor(mx, off, 32));
    float s = expf(a - mx) + expf(b - mx);
#pragma unroll
    for (int off = 16; off > 0; off >>= 1)
        s += __shfl_xor(s, off, 32);
    float lse = mx + logf(s);
    p[lane]      = a - lse;
    p[lane + 32] = b - lse;
}

extern "C" void kernel_launch(void* const* d_in, const int* in_sizes, int n_in,
                              void* d_out, int out_size, void* d_ws, size_t ws_size,
                              hipStream_t stream) {
    const float* x    = (const float*)d_in[0];
    const int*   src1 = (const int*)d_in[1];
    const int*   dst1 = (const int*)d_in[2];
    const int*   src2 = (const int*)d_in[3];
    const int*   dst2 = (const int*)d_in[4];
    const float* w_l1 = (const float*)d_in[5];
    const float* b_l1 = (const float*)d_in[6];
    const float* w_r1 = (const float*)d_in[7];
    const float* w_l2 = (const float*)d_in[8];
    const float* b_l2 = (const float*)d_in[9];
    const float* w_r2 = (const float*)d_in[10];
    const int E1 = in_sizes[1];
    const int E2 = in_sizes[3];
    float* out = (float*)d_out;

    // Workspace carve-up (~66 MB total)
    float* agg1 = (float*)d_ws;                    // [N1, IN]
    float* deg1 = agg1 + (size_t)N1 * IN;          // [N1]
    float* h    = deg1 + N1;                       // [N1, HID]
    float* agg2 = h    + (size_t)N1 * HID;         // [N2, HID]
    float* deg2 = agg2 + (size_t)N2 * HID;         // [N2]

    // Zero the accumulators (contiguous regions; graph-capturable)
    hipMemsetAsync(agg1, 0, ((size_t)N1 * IN  + N1) * sizeof(float), stream);
    hipMemsetAsync(agg2, 0, ((size_t)N2 * HID + N2) * sizeof(float), stream);

    // ---- Layer 1 ----
    {   // scatter-add x[src1] -> agg1[dst1], count degrees
        long long threads = (long long)E1 * 32;
        sage_scatter<IN><<<dim3((unsigned)((threads + 255) / 256)), dim3(256), 0, stream>>>(
            x, src1, dst1, agg1, deg1, E1);
    }
    {   // mean
        int total = N1 * IN;
        sage_norm<<<dim3((total + 255) / 256), dim3(256), 0, stream>>>(agg1, deg1, total, 7);
    }
    {   // h = relu(agg1 @ w_l1 + b_l1 + x[:N1] @ w_r1)   (WMMA f32)
        int waves = (N1 / 16) * (HID / (16 * 4));          // 2500 * 4 = 10000
        sage_gemm<IN, HID, 4, true><<<dim3((waves * 32 + 255) / 256), dim3(256), 0, stream>>>(
            agg1, x, w_l1, w_r1, b_l1, h, N1 / 16);
    }

    // ---- Layer 2 ----
    {
        long long threads = (long long)E2 * 32;
        sage_scatter<HID><<<dim3((unsigned)((threads + 255) / 256)), dim3(256), 0, stream>>>(
            h, src2, dst2, agg2, deg2, E2);
    }
    {
        int total = N2 * HID;
        sage_norm<<<dim3((total + 255) / 256), dim3(256), 0, stream>>>(agg2, deg2, total, 8);
    }
    {   // out = agg2 @ w_l2 + b_l2 + h[:N2] @ w_r2   (WMMA f32, no relu)
        int waves = (N2 / 16) * (OUT / 16);                // 250 * 4 = 1000
        sage_gemm<HID, OUT, 1, false><<<dim3((waves * 32 + 255) / 256), dim3(256), 0, stream>>>(
            agg2, h, w_l2, w_r2, b_l2, out, N2 / 16);
    }
    {   // in-place log_softmax on d_out
        int waves = N2;
        sage_logsoftmax<<<dim3((waves * 32 + 255) / 256), dim3(256), 0, stream>>>(out, N2);
    }
}